// HyboNetnoEncoder_68633577390231
// MI455X (gfx1250) — compile-verified
//
#include <hip/hip_runtime.h>
#include <math.h>

// CDNA5 f32 WMMA fragment types: A/B = 16x4 / 4x16 f32 -> 2 VGPRs, C/D = 16x16 f32 -> 8 VGPRs
typedef __attribute__((ext_vector_type(2))) float v2f;
typedef __attribute__((ext_vector_type(8))) float v8f;

// Only meaningful during the device pass; host clang never has amdgcn builtins.
#if defined(__HIP_DEVICE_COMPILE__)
#if !__has_builtin(__builtin_amdgcn_wmma_f32_16x16x4_f32)
#error "device pass: missing __builtin_amdgcn_wmma_f32_16x16x4_f32"
#endif
#endif

#define DIMX 128
#define WPAD 132          // 128 + 4 pad floats -> row stride 528B, bank-conflict free
#define LIST_MAX 8192

__global__ __launch_bounds__(256)
void hybonet_kernel(const int* __restrict__ u_idx,
                    const int* __restrict__ r_idx,
                    const int* __restrict__ v_idx,
                    const float* __restrict__ drug,
                    const float* __restrict__ target,
                    const float* __restrict__ rbias,
                    const float* __restrict__ rtrans,
                    const float* __restrict__ bias_head,
                    const float* __restrict__ bias_tail,
                    float* __restrict__ out,
                    int B, int K)
{
    __shared__ float Wl[DIMX * WPAD];        // W_r, padded rows (67.6 KB)
    __shared__ float Ht2[DIMX * 16];         // 16 head columns, (d-pair, col) interleaved (8 KB)
    __shared__ float Xl[16 * WPAD];          // X tile, column-major per batch col (8.4 KB)
    __shared__ float rbl[DIMX];              // relation bias row
    __shared__ unsigned short list[LIST_MAX];// batch rows belonging to this relation
    __shared__ int cnt;

    const int r    = blockIdx.x;
    const int tid  = threadIdx.x;
    const int lane = tid & 31;
    const int wv   = tid >> 5;               // wave id 0..7 (wave32)

    if (tid == 0) cnt = 0;
    __syncthreads();

    // ---- group batch rows by relation: scan r_idx, collect matches ----
    for (int i = tid; i < B; i += 256) {
        if (r_idx[i] == r) {
            int p = atomicAdd(&cnt, 1);
            if (p < LIST_MAX) list[p] = (unsigned short)i;
        }
    }

    // ---- stage W_r (row-major e x d) into padded LDS, and rb_r ----
    {
        const float* Wg = rtrans + (size_t)r * DIMX * DIMX;
        for (int i = tid; i < DIMX * (DIMX / 4); i += 256) {
            int row = i >> 5;
            int c4  = (i & 31) << 2;
            float4 wq = *(const float4*)(Wg + row * DIMX + c4);
            *(float4*)&Wl[row * WPAD + c4] = wq;
        }
        if (tid < DIMX) rbl[tid] = rbias[r * DIMX + tid];
    }
    __syncthreads();

    const int n = (cnt < LIST_MAX) ? cnt : LIST_MAX;

    for (int t0 = 0; t0 < n; t0 += 16) {
        // ---- gather up to 16 head columns into interleaved LDS buffer ----
        {
            int col  = tid >> 4;             // 0..15
            int dl   = tid & 15;
            int cidx = t0 + col;
            const float* hp = nullptr;
            if (cidx < n) hp = drug + (size_t)u_idx[list[cidx]] * DIMX;
            for (int j = 0; j < 8; j++) {
                int d = dl + (j << 4);
                float hv = hp ? hp[d] : 0.0f;
                // pairs (d even, d+1) adjacent per column -> 8B v2f loads later
                Ht2[((d >> 1) << 5) + (col << 1) + (d & 1)] = hv;
            }
        }
        __syncthreads();

        // ---- X(128x16) = W_r(128x128) * H(128x16) via V_WMMA_F32_16X16X4_F32 ----
        // wave wv owns rows [wv*16, wv*16+16); EXEC all-ones, uniform control flow.
        {
            const int m  = lane & 15;
            const int hi = lane >> 4;        // lanes 16-31 carry K+2 (ISA 16x4 f32 A layout)
            const float* Arow = &Wl[(wv * 16 + m) * WPAD];
            v8f c = {0.f, 0.f, 0.f, 0.f, 0.f, 0.f, 0.f, 0.f};
            for (int kc = 0; kc < DIMX; kc += 4) {
                v2f a  = *(const v2f*)(Arow + kc + hi * 2);                       // A[m, kc+hi*2 .. +1]
                v2f bb = *(const v2f*)(&Ht2[(((kc + hi * 2) >> 1) << 5) + (m << 1)]); // B[kc+hi*2.., col=m]
                c = __builtin_amdgcn_wmma_f32_16x16x4_f32(false, a, false, bb,
                                                          (short)0, c, false, false);
            }
            // C/D layout: VGPR i -> row (i + hi*8), col = lane&15
            for (int i = 0; i < 8; i++)
                Xl[m * WPAD + wv * 16 + hi * 8 + i] = c[i];
        }
        __syncthreads();

        // ---- per-column nonlinearity + K tail dots (wave wv -> cols 2wv, 2wv+1) ----
        for (int cc = 0; cc < 2; cc++) {
            int col  = wv * 2 + cc;
            int cidx = t0 + col;
            if (cidx < n) {
                int b = list[cidx];
                const float x0 = Xl[col * WPAD];                 // pre-bias x[0]
                const float tm = 2.5f / (1.0f + __expf(-x0)) + 1.1f;
                const int e0 = lane << 2;
                float zz[4];
                float ss = 0.0f;
                for (int j = 0; j < 4; j++) {
                    int e = e0 + j;
                    float nv = Xl[col * WPAD + e] + rbl[e];
                    zz[j] = nv;
                    if (e >= 1) ss += nv * nv;
                }
                for (int msk = 16; msk >= 1; msk >>= 1) ss += __shfl_xor(ss, msk, 32);
                const float scale = __fsqrt_rn((tm * tm - 1.0f) / ss);
                for (int j = 0; j < 4; j++) zz[j] *= scale;
                if (lane == 0) zz[0] = -tm;                      // fold -time*tail[0] into the dot
                const float bh = bias_head[u_idx[b]];
                const int* vr = v_idx + (size_t)b * K;
                float keep = 0.0f;
                for (int k = 0; k < K; k++) {
                    int vt = vr[k];
                    const float4 t4 = *(const float4*)(target + (size_t)vt * DIMX + e0);
                    float p = zz[0] * t4.x + zz[1] * t4.y + zz[2] * t4.z + zz[3] * t4.w;
                    for (int msk = 16; msk >= 1; msk >>= 1) p += __shfl_xor(p, msk, 32);
                    float res = 8.0f + 2.0f * p + bh + bias_tail[vt];
                    if (lane == (k & 31)) keep = res;
                    if ((k & 31) == 31)                          // coalesced 32-wide store
                        out[(size_t)b * K + (k - 31) + lane] = keep;
                }
                int rem = K & 31;
                if (rem && lane < rem) out[(size_t)b * K + (K - rem) + lane] = keep;
            }
        }
        __syncthreads();
    }
}

extern "C" void kernel_launch(void* const* d_in, const int* in_sizes, int n_in,
                              void* d_out, int out_size, void* d_ws, size_t ws_size,
                              hipStream_t stream) {
    const int*   u_idx  = (const int*)d_in[0];
    const int*   r_idx  = (const int*)d_in[1];
    const int*   v_idx  = (const int*)d_in[2];
    const float* drug   = (const float*)d_in[3];
    const float* target = (const float*)d_in[4];
    const float* rbias  = (const float*)d_in[5];
    const float* rtrans = (const float*)d_in[6];
    const float* bh     = (const float*)d_in[7];
    const float* bt     = (const float*)d_in[8];
    float* out = (float*)d_out;

    const int B    = in_sizes[0];
    const int K    = in_sizes[2] / B;
    const int nrel = in_sizes[5] / DIMX;

    hybonet_kernel<<<dim3(nrel), dim3(256), 0, stream>>>(
        u_idx, r_idx, v_idx, drug, target, rbias, rtrans, bh, bt, out, B, K);
}